// GATEncoder_9131100471789
// MI455X (gfx1250) — compile-verified
//
#include <hip/hip_runtime.h>
#include <math.h>

// Problem constants (match reference)
#define NNODES   20000
#define NEDGES   320000
#define INC      128
#define HIDC     64
#define NHEADS   4
#define EDIMC    16
#define NEG_SLOPE 0.2f

typedef __attribute__((ext_vector_type(2))) float v2f;
typedef __attribute__((ext_vector_type(8))) float v8f;

// ---------------------------------------------------------------------------
// WMMA fp32 GEMM:  C[M,N] = A[M,K] @ B[K,N]   (row-major, M%16==0, N%64==0,
// K%4==0).  Block = 4 waves = 64 N-columns; the K x 64 B-panel is staged in
// LDS once and reused across GEMM_MT M-tiles (8x fewer B fetches).  Each wave
// computes 16x16 C tiles with a K-loop of V_WMMA_F32_16X16X4_F32.
// Fragment layouts per CDNA5 ISA 7.12.2:
//   A 16x4 : lane holds row M=lane&15; element v <-> K = k0 + v + 2*(lane>>4)
//   B 4x16 : lane holds col N=lane&15; element v <-> K = k0 + v + 2*(lane>>4)
//   C 16x16: VGPR r <-> row M = r + 8*(lane>>4), col N = lane&15
// ---------------------------------------------------------------------------
#define GEMM_WAVES 4
#define GEMM_MT    8
__global__ __launch_bounds__(32 * GEMM_WAVES)
void wmma_gemm_f32(const float* __restrict__ A,
                   const float* __restrict__ B,
                   float* __restrict__ C,
                   int M, int Ncols, int K) {
  extern __shared__ float bpanel[];            // [K][64] floats
  const int lane = threadIdx.x & 31;
  const int wave = threadIdx.x >> 5;
  const int nblk = blockIdx.y * (16 * GEMM_WAVES);   // first column of block

  // cooperative stage of B[0:K, nblk:nblk+64) into LDS
  for (int i = threadIdx.x; i < K * 64; i += blockDim.x) {
    int k = i >> 6;
    int n = i & 63;
    bpanel[i] = B[(size_t)k * Ncols + nblk + n];
  }
  __syncthreads();

  const int sub   = lane >> 4;                 // lane half: 0 or 1
  const int khalf = sub * 2;
  const int colL  = wave * 16 + (lane & 15);   // column within LDS panel
  const int col   = nblk + colL;               // global column

  for (int t = 0; t < GEMM_MT; ++t) {
    const int m0 = (blockIdx.x * GEMM_MT + t) * 16;
    if (m0 >= M) break;
    const float* arow = A + (size_t)(m0 + (lane & 15)) * K;

    v8f acc = {};
    for (int k0 = 0; k0 < K; k0 += 4) {
      v2f a, b;
      a.x = arow[k0 + khalf + 0];
      a.y = arow[k0 + khalf + 1];
      b.x = bpanel[(k0 + khalf + 0) * 64 + colL];
      b.y = bpanel[(k0 + khalf + 1) * 64 + colL];
      acc = __builtin_amdgcn_wmma_f32_16x16x4_f32(
          /*neg_a=*/false, a, /*neg_b=*/false, b,
          /*c_mod=*/(short)0, acc, /*reuse_a=*/false, /*reuse_b=*/false);
    }
#pragma unroll
    for (int r = 0; r < 8; ++r) {
      C[(size_t)(m0 + r + 8 * sub) * Ncols + col] = acc[r];
    }
  }
}

// ---------------------------------------------------------------------------
__global__ void fill_f32(float* __restrict__ p, float v, size_t n) {
  size_t i = (size_t)blockIdx.x * blockDim.x + threadIdx.x;
  size_t stride = (size_t)gridDim.x * blockDim.x;
  for (; i < n; i += stride) p[i] = v;
}

// float atomic max via signed/unsigned int ordering trick (valid for all
// sign combinations given -inf initialization).
__device__ __forceinline__ void atomicMaxF(float* addr, float val) {
  if (val >= 0.0f) {
    atomicMax((int*)addr, __float_as_int(val));
  } else {
    atomicMin((unsigned int*)addr, __float_as_uint(val));
  }
}

// ---------------------------------------------------------------------------
// Per-(edge,head): logit = sum_c att[h,c]*leakyrelu(xl[s]+xr[d]+ea).  The
// edge-attr projection ea (K=16) is contracted on the fly (never materialize
// the E x H*C tensor); We/att are staged in LDS (broadcast reads).  Segment
// max folded in via float atomic-max.
// ---------------------------------------------------------------------------
__global__ void edge_logits_kernel(const float* __restrict__ xl,
                                   const float* __restrict__ xr,
                                   const float* __restrict__ eattr, // [E,16]
                                   const float* __restrict__ We,    // [16,H*C]
                                   const float* __restrict__ att,   // [H,C]
                                   const int* __restrict__ src,
                                   const int* __restrict__ dst,
                                   float* __restrict__ logits,      // [E,H]
                                   float* __restrict__ mbuf,        // [N,H]
                                   int H, int C) {
  extern __shared__ float sm[];                // [16*H*C + H*C]
  const int HC = H * C;
  float* sWe  = sm;                            // [16, H*C]
  float* satt = sm + EDIMC * HC;               // [H, C]
  for (int i = threadIdx.x; i < EDIMC * HC; i += blockDim.x) sWe[i] = We[i];
  for (int i = threadIdx.x; i < HC; i += blockDim.x) satt[i] = att[i];
  __syncthreads();

  int tid = blockIdx.x * blockDim.x + threadIdx.x;
  int total = NEDGES * H;
  if (tid >= total) return;
  int e = tid / H;
  int h = tid - e * H;
  int s = src[e], d = dst[e];

  const float* xls = xl + (size_t)s * HC + h * C;
  const float* xrd = xr + (size_t)d * HC + h * C;
  __builtin_prefetch(xls, 0, 0);               // global_prefetch_b8
  __builtin_prefetch(xrd, 0, 0);
  const float* ap = satt + h * C;
  const float* Wh = sWe + h * C;

  float ea[EDIMC];
#pragma unroll
  for (int k = 0; k < EDIMC; ++k) ea[k] = eattr[(size_t)e * EDIMC + k];

  float logit = 0.0f;
  for (int c = 0; c < C; ++c) {
    float v = xls[c] + xrd[c];
#pragma unroll
    for (int k = 0; k < EDIMC; ++k) v += ea[k] * Wh[k * HC + c];
    v = (v > 0.0f) ? v : NEG_SLOPE * v;        // LeakyReLU(0.2)
    logit += ap[c] * v;
  }
  logits[tid] = logit;
  atomicMaxF(&mbuf[(size_t)d * H + h], logit);
}

// Per-(edge,head): alpha = exp(logit - m[dst]); accumulate softmax denom.
__global__ void edge_exp_kernel(const float* __restrict__ logits,
                                const float* __restrict__ mbuf,
                                const int* __restrict__ dst,
                                float* __restrict__ alpha,
                                float* __restrict__ denom, int H) {
  int tid = blockIdx.x * blockDim.x + threadIdx.x;
  int total = NEDGES * H;
  if (tid >= total) return;
  int e = tid / H;
  int h = tid - e * H;
  int d = dst[e];
  float a = expf(logits[tid] - mbuf[(size_t)d * H + h]);
  alpha[tid] = a;
  atomicAdd(&denom[(size_t)d * H + h], a);
}

// Per-(edge,head,channel): acc[dst] += (alpha/denom) * xl[src]   (GATv2
// message is the source-side projection).
__global__ void edge_scatter_kernel(const float* __restrict__ alpha,
                                    const float* __restrict__ denom,
                                    const float* __restrict__ xl,
                                    const int* __restrict__ src,
                                    const int* __restrict__ dst,
                                    float* __restrict__ acc,
                                    int H, int C) {
  size_t total = (size_t)NEDGES * H * C;
  size_t i = (size_t)blockIdx.x * blockDim.x + threadIdx.x;
  size_t stride = (size_t)gridDim.x * blockDim.x;
  for (; i < total; i += stride) {
    int    c  = (int)(i % C);
    size_t eh = i / C;
    int    h  = (int)(eh % H);
    int    e  = (int)(eh / H);
    int s = src[e], d = dst[e];
    float w = alpha[eh] / denom[(size_t)d * H + h];
    atomicAdd(&acc[(size_t)d * H * C + h * C + c],
              w * xl[(size_t)s * H * C + h * C + c]);
  }
}

__global__ void bias_elu_kernel(float* __restrict__ h,
                                const float* __restrict__ b,
                                size_t n, int HC) {
  size_t i = (size_t)blockIdx.x * blockDim.x + threadIdx.x;
  size_t stride = (size_t)gridDim.x * blockDim.x;
  for (; i < n; i += stride) {
    float v = h[i] + b[i % HC];
    h[i] = (v > 0.0f) ? v : expm1f(v);         // ELU(alpha=1)
  }
}

__global__ void bias_add_kernel(float* __restrict__ o,
                                const float* __restrict__ b,
                                size_t n, int HC) {
  size_t i = (size_t)blockIdx.x * blockDim.x + threadIdx.x;
  size_t stride = (size_t)gridDim.x * blockDim.x;
  for (; i < n; i += stride) o[i] += b[i % HC];
}

// ---------------------------------------------------------------------------
extern "C" void kernel_launch(void* const* d_in, const int* in_sizes, int n_in,
                              void* d_out, int out_size, void* d_ws, size_t ws_size,
                              hipStream_t stream) {
  (void)in_sizes; (void)n_in; (void)out_size; (void)ws_size;

  const float* x     = (const float*)d_in[0];
  const int*   ei    = (const int*)  d_in[1];
  const float* eattr = (const float*)d_in[2];
  const float* Wl1   = (const float*)d_in[3];
  const float* Wr1   = (const float*)d_in[4];
  const float* We1   = (const float*)d_in[5];
  const float* att1  = (const float*)d_in[6];
  const float* b1    = (const float*)d_in[7];
  const float* Wl2   = (const float*)d_in[8];
  const float* Wr2   = (const float*)d_in[9];
  const float* We2   = (const float*)d_in[10];
  const float* att2  = (const float*)d_in[11];
  const float* b2    = (const float*)d_in[12];
  float* out = (float*)d_out;
  float* ws  = (float*)d_ws;

  const int* src = ei;            // edge_index[0]
  const int* dst = ei + NEDGES;   // edge_index[1]

  const int HC1 = NHEADS * HIDC;  // 256

  // workspace layout (floats)
  size_t o = 0;
  float* xl1    = ws + o; o += (size_t)NNODES * HC1;
  float* xr1    = ws + o; o += (size_t)NNODES * HC1;
  float* hbuf   = ws + o; o += (size_t)NNODES * HC1;   // acc1 -> elu(h)
  float* xl2    = ws + o; o += (size_t)NNODES * HIDC;
  float* xr2    = ws + o; o += (size_t)NNODES * HIDC;
  float* logits = ws + o; o += (size_t)NEDGES * NHEADS;
  float* alpha  = ws + o; o += (size_t)NEDGES * NHEADS;
  float* mbuf   = ws + o; o += (size_t)NNODES * NHEADS;
  float* denom  = ws + o; o += (size_t)NNODES * NHEADS;

  dim3 gblk(32 * GEMM_WAVES);
  const int mtiles  = NNODES / 16;                         // 1250
  const int gx      = (mtiles + GEMM_MT - 1) / GEMM_MT;    // 157

  // ---- layer 1 node projections (WMMA fp32, LDS-staged B panel) ----
  dim3 g1(gx, HC1 / (16 * GEMM_WAVES));                    // 157 x 4
  size_t shm1 = (size_t)INC * 64 * sizeof(float);          // 32 KB
  wmma_gemm_f32<<<g1, gblk, shm1, stream>>>(x, Wl1, xl1, NNODES, HC1, INC);
  wmma_gemm_f32<<<g1, gblk, shm1, stream>>>(x, Wr1, xr1, NNODES, HC1, INC);

  // ---- layer 1 attention / aggregation ----
  fill_f32<<<512, 256, 0, stream>>>(mbuf, -INFINITY, (size_t)NNODES * NHEADS);
  fill_f32<<<512, 256, 0, stream>>>(denom, 0.0f, (size_t)NNODES * NHEADS);
  fill_f32<<<2048, 256, 0, stream>>>(hbuf, 0.0f, (size_t)NNODES * HC1);

  int eh1 = NEDGES * NHEADS;
  size_t eshm1 = (size_t)(EDIMC * HC1 + HC1) * sizeof(float);  // ~17 KB
  edge_logits_kernel<<<(eh1 + 255) / 256, 256, eshm1, stream>>>(
      xl1, xr1, eattr, We1, att1, src, dst, logits, mbuf, NHEADS, HIDC);
  edge_exp_kernel<<<(eh1 + 255) / 256, 256, 0, stream>>>(
      logits, mbuf, dst, alpha, denom, NHEADS);
  edge_scatter_kernel<<<8192, 256, 0, stream>>>(
      alpha, denom, xl1, src, dst, hbuf, NHEADS, HIDC);
  bias_elu_kernel<<<2048, 256, 0, stream>>>(hbuf, b1, (size_t)NNODES * HC1, HC1);

  // ---- layer 2 node projections (WMMA fp32, K=256, N=64) ----
  dim3 g2(gx, HIDC / (16 * GEMM_WAVES));                   // 157 x 1
  size_t shm2 = (size_t)HC1 * 64 * sizeof(float);          // 64 KB
  wmma_gemm_f32<<<g2, gblk, shm2, stream>>>(hbuf, Wl2, xl2, NNODES, HIDC, HC1);
  wmma_gemm_f32<<<g2, gblk, shm2, stream>>>(hbuf, Wr2, xr2, NNODES, HIDC, HC1);

  // ---- layer 2 attention / aggregation (H=1, accumulate into d_out) ----
  fill_f32<<<512, 256, 0, stream>>>(mbuf, -INFINITY, (size_t)NNODES);
  fill_f32<<<512, 256, 0, stream>>>(denom, 0.0f, (size_t)NNODES);
  fill_f32<<<512, 256, 0, stream>>>(out, 0.0f, (size_t)NNODES * HIDC);

  size_t eshm2 = (size_t)(EDIMC * HIDC + HIDC) * sizeof(float);  // ~4 KB
  edge_logits_kernel<<<(NEDGES + 255) / 256, 256, eshm2, stream>>>(
      xl2, xr2, eattr, We2, att2, src, dst, logits, mbuf, 1, HIDC);
  edge_exp_kernel<<<(NEDGES + 255) / 256, 256, 0, stream>>>(
      logits, mbuf, dst, alpha, denom, 1);
  edge_scatter_kernel<<<8192, 256, 0, stream>>>(
      alpha, denom, xl2, src, dst, out, 1, HIDC);
  bias_add_kernel<<<512, 256, 0, stream>>>(out, b2, (size_t)NNODES * HIDC, HIDC);
}